// SimpleRetention_10488310136848
// MI455X (gfx1250) — compile-verified
//
#include <hip/hip_runtime.h>
#include <hip/hip_bf16.h>

// SimpleRetention forward for MI455X (gfx1250), f32 WMMA (16x16x4) throughout.
// B=32, S=1024, E=512, H=64, N_AGENTS=4, T=256, kappa=0.9.

typedef float v2f __attribute__((ext_vector_type(2)));
typedef float v8f __attribute__((ext_vector_type(8)));

constexpr int   Bc = 32, Sc = 1024, Ec = 512, Hc = 64, Tc = 256;
constexpr float LNK = -0.1053605156578263f;   // ln(0.9)

static __device__ __forceinline__ v8f wmma4(v2f a, v2f b, v8f c) {
  // D = A(16x4,f32) x B(4x16,f32) + C(16x16,f32)
  return __builtin_amdgcn_wmma_f32_16x16x4_f32(false, a, false, b, (short)0, c,
                                               false, false);
}

// ---------------------------------------------------------------------------
// Y(MxN) = X(MxK) @ W(KxN), one wave per 16x16 output tile, K in steps of 4.
// A-layout: lane m = lane%16; VGPR0 holds K=khalf, VGPR1 K=khalf+1 (khalf = 0
// for lanes 0-15, 2 for lanes 16-31) -> contiguous v2f load along K.
// B-layout: lane n = lane%16; VGPR0 row K=khalf, VGPR1 row K=khalf+1.
// ---------------------------------------------------------------------------
__global__ void __launch_bounds__(32)
proj16x16(const float* __restrict__ X, const float* __restrict__ W,
          float* __restrict__ Y, int K, int N)
{
  const int lane  = threadIdx.x & 31;
  const int nl    = lane & 15;
  const int khalf = (lane >> 4) << 1;
  const int m     = (blockIdx.x << 4) + nl;
  const int ncol  = (blockIdx.y << 4) + nl;

  const float* xrow = X + (size_t)m * K;
  v8f c = {};
  for (int k0 = 0; k0 < K; k0 += 4) {
    v2f a = *(const v2f*)(xrow + k0 + khalf);
    v2f b;
    b.x = W[(size_t)(k0 + khalf)     * N + ncol];
    b.y = W[(size_t)(k0 + khalf + 1) * N + ncol];
    c = wmma4(a, b, c);
  }
  // C/D layout: VGPR r, lane l -> M = r + 8*(l>=16), N = l%16
  const int mbase = (blockIdx.x << 4) + ((lane >> 4) << 3);
  #pragma unroll
  for (int r = 0; r < 8; ++r)
    Y[(size_t)(mbase + r) * N + ncol] = c[r];
}

// ---------------------------------------------------------------------------
// Per-batch inclusive cumsum of timestep dones + first-done timestep.
// ---------------------------------------------------------------------------
__global__ void dones_scan(const int* __restrict__ dones, int* __restrict__ cum,
                           int* __restrict__ firstdone)
{
  const int b = blockIdx.x;
  if (threadIdx.x == 0) {
    int c = 0, first = Tc;
    for (int t = 0; t < Tc; ++t) {
      const int d = (dones[(size_t)b * Sc + t * 4] != 0) ? 1 : 0;
      if (d && first == Tc) first = t;
      c += d;
      cum[b * Tc + t] = c;
    }
    firstdone[b] = first;
  }
}

// ---------------------------------------------------------------------------
// out[b, i0:i0+16, :] = sum_{jt<=it} (q@k^T * dm)@v  +  (q@hstate) * xi
// One wave per (row-strip of 16, batch). The q tile (16x64) is held in
// registers as 16 A-fragments for the whole wave. Masked 16x16 score tile
// round-trips LDS to convert C-layout -> A-layout for the second WMMA.
// Decay factors and done-cumsum live in LDS.
// ---------------------------------------------------------------------------
__global__ void __launch_bounds__(32)
retention16(const float* __restrict__ q, const float* __restrict__ kbuf,
            const float* __restrict__ vbuf, const float* __restrict__ hstate,
            const int* __restrict__ cum, const int* __restrict__ firstdone,
            float* __restrict__ out)
{
  __shared__ float ktab[Tc + 1];
  __shared__ int   cumsh[Tc];
  __shared__ float atile[16 * 16];

  const int lane  = threadIdx.x & 31;
  const int nl    = lane & 15;
  const int khalf = (lane >> 4) << 1;
  const int mh    = (lane >> 4) << 3;
  const int it    = blockIdx.x;       // row tile (16 rows)
  const int b     = blockIdx.y;

  const int* cumb = cum + b * Tc;
  for (int i = lane; i <= Tc; i += 32) ktab[i] = __expf((float)i * LNK);
  for (int i = lane; i < Tc; i += 32)  cumsh[i] = cumb[i];
  __syncthreads();

  const float* qb = q      + (size_t)b * Sc * Hc;
  const float* kb = kbuf   + (size_t)b * Sc * Hc;
  const float* vb = vbuf   + (size_t)b * Sc * Hc;
  const float* hb = hstate + (size_t)b * Hc * Hc;

  // ---- hoist q tile into registers: 16 A-fragments (32 VGPRs/lane) ----
  const float* qrow = qb + (size_t)(it * 16 + nl) * Hc;
  v2f qa[16];
  #pragma unroll
  for (int ks = 0; ks < 16; ++ks)
    qa[ks] = *(const v2f*)(qrow + ks * 4 + khalf);

  v8f acc[4] = {};                    // 16 rows x 64 cols of output
  for (int jt = 0; jt <= it; ++jt) {
    // ---- s = q_tile @ k_tile^T  (K = H = 64) ----
    v8f s = {};
    const float* krow = kb + (size_t)(jt * 16 + nl) * Hc;  // B[K][n]=k[n][K]
    #pragma unroll
    for (int ks = 0; ks < 16; ++ks) {
      v2f bm = *(const v2f*)(krow + ks * 4 + khalf);
      s = wmma4(qa[ks], bm, s);
    }
    // ---- apply decay mask dm(gi,gj), stage to LDS ----
    #pragma unroll
    for (int r = 0; r < 8; ++r) {
      const int gi = it * 16 + mh + r;
      const int gj = jt * 16 + nl;
      const int ti = gi >> 2, tj = gj >> 2;
      float w = 0.0f;
      if (gj <= gi && cumsh[ti] == cumsh[tj]) w = ktab[ti - tj];
      atile[(mh + r) * 16 + nl] = s[r] * w;
    }
    __syncthreads();
    // ---- acc += atile(16x16) @ v_tile(16x64)  (K = 16) ----
    #pragma unroll
    for (int k0 = 0; k0 < 16; k0 += 4) {
      v2f a = *(const v2f*)(&atile[nl * 16 + k0 + khalf]);
      const float* v0 = vb + (size_t)(jt * 16 + k0 + khalf) * Hc;
      const float* v1 = v0 + Hc;
      #pragma unroll
      for (int nt = 0; nt < 4; ++nt) {
        v2f bm;
        bm.x = v0[nt * 16 + nl];
        bm.y = v1[nt * 16 + nl];
        acc[nt] = wmma4(a, bm, acc[nt]);
      }
    }
    __syncthreads();
  }

  // ---- cross chunk: (q_tile @ hstate) * xi(row) ----
  v8f cr[4] = {};
  #pragma unroll
  for (int ks = 0; ks < 16; ++ks) {
    const float* h0 = hb + (size_t)(ks * 4 + khalf) * Hc;
    const float* h1 = h0 + Hc;
    #pragma unroll
    for (int nt = 0; nt < 4; ++nt) {
      v2f bm;
      bm.x = h0[nt * 16 + nl];
      bm.y = h1[nt * 16 + nl];
      cr[nt] = wmma4(qa[ks], bm, cr[nt]);
    }
  }

  const int fd = firstdone[b];
  #pragma unroll
  for (int r = 0; r < 8; ++r) {
    const int   gi = it * 16 + mh + r;
    const int   ti = gi >> 2;
    const float xi = (ti < fd) ? ktab[ti + 1] : 0.0f;
    float* orow = out + ((size_t)b * Sc + gi) * Hc;
    #pragma unroll
    for (int nt = 0; nt < 4; ++nt)
      orow[nt * 16 + nl] = acc[nt][r] + cr[nt][r] * xi;
  }
}

// ---------------------------------------------------------------------------
// next_hstate = k^T @ (v * dm_last) + hstate * kappa^T * delta
// One wave per (16x16 tile of 64x64 state, batch), K = S = 1024.
// ---------------------------------------------------------------------------
__global__ void __launch_bounds__(32)
hstate_update(const float* __restrict__ kbuf, const float* __restrict__ vbuf,
              const float* __restrict__ hstate, const int* __restrict__ cum,
              float* __restrict__ hout)
{
  __shared__ float ktab[Tc + 1];
  __shared__ float wsh[Tc];           // per-timestep weight of dm's last row
  const int lane  = threadIdx.x & 31;
  const int nl    = lane & 15;
  const int khalf = (lane >> 4) << 1;
  const int mt = blockIdx.x, nt = blockIdx.y, b = blockIdx.z;
  const int* cumb = cum + b * Tc;
  const int  cend = cumb[Tc - 1];

  for (int i = lane; i <= Tc; i += 32) ktab[i] = __expf((float)i * LNK);
  __syncthreads();
  for (int i = lane; i < Tc; i += 32)
    wsh[i] = (cumb[i] == cend) ? ktab[(Tc - 1) - i] : 0.0f;
  __syncthreads();

  const float* kb = kbuf + (size_t)b * Sc * Hc;
  const float* vb = vbuf + (size_t)b * Sc * Hc;

  v8f c = {};
  for (int s0 = 0; s0 < Sc; s0 += 4) {
    const int   sA = s0 + khalf;       // sA, sA+1 are in the same timestep
    const float w  = wsh[s0 >> 2];
    v2f a;                             // A[m][s] = k[s][m]
    a.x = kb[(size_t)sA       * Hc + mt * 16 + nl];
    a.y = kb[(size_t)(sA + 1) * Hc + mt * 16 + nl];
    v2f bm;                            // B[s][n] = v[s][n] * w
    bm.x = vb[(size_t)sA       * Hc + nt * 16 + nl] * w;
    bm.y = vb[(size_t)(sA + 1) * Hc + nt * 16 + nl] * w;
    c = wmma4(a, bm, c);
  }

  const float cd    = ktab[Tc];                 // kappa^256
  const float delta = (cend == 0) ? 1.0f : 0.0f;
  const float* hb   = hstate + (size_t)b * Hc * Hc;
  const int    mh   = (lane >> 4) << 3;
  #pragma unroll
  for (int r = 0; r < 8; ++r) {
    const int h1 = mt * 16 + mh + r;
    hout[((size_t)b * Hc + h1) * Hc + nt * 16 + nl] =
        c[r] + hb[(size_t)h1 * Hc + nt * 16 + nl] * cd * delta;
  }
}

// ---------------------------------------------------------------------------
extern "C" void kernel_launch(void* const* d_in, const int* in_sizes, int n_in,
                              void* d_out, int out_size, void* d_ws, size_t ws_size,
                              hipStream_t stream)
{
  const float* key    = (const float*)d_in[0];
  const float* query  = (const float*)d_in[1];
  const float* value  = (const float*)d_in[2];
  const float* hstate = (const float*)d_in[3];
  const int*   dones  = (const int*)d_in[4];
  const float* W_Q    = (const float*)d_in[5];
  const float* W_K    = (const float*)d_in[6];
  const float* W_V    = (const float*)d_in[7];

  const size_t qkv = (size_t)Bc * Sc * Hc;   // 2,097,152 floats each
  float* ws = (float*)d_ws;
  float* qb = ws;
  float* kb = ws + qkv;
  float* vb = ws + 2 * qkv;
  int*   cum       = (int*)(ws + 3 * qkv);
  int*   firstdone = cum + Bc * Tc;

  dim3 gproj(Bc * Sc / 16, Hc / 16);
  proj16x16<<<gproj, 32, 0, stream>>>(query, W_Q, qb, Ec, Hc);
  proj16x16<<<gproj, 32, 0, stream>>>(key,   W_K, kb, Ec, Hc);
  proj16x16<<<gproj, 32, 0, stream>>>(value, W_V, vb, Ec, Hc);

  dones_scan<<<Bc, 32, 0, stream>>>(dones, cum, firstdone);

  float* out0 = (float*)d_out;
  float* hout = out0 + qkv;
  retention16<<<dim3(Sc / 16, Bc), 32, 0, stream>>>(qb, kb, vb, hstate, cum,
                                                    firstdone, out0);
  hstate_update<<<dim3(Hc / 16, Hc / 16, Bc), 32, 0, stream>>>(kb, vb, hstate,
                                                               cum, hout);
}